// IF_31636729102650
// MI455X (gfx1250) — compile-verified
//
#include <hip/hip_runtime.h>

// IF (integrate-and-fire) forward, T=4, pure streaming kernel.
// Memory-bound: 268 MB moved, ~11.5 us floor at 23.3 TB/s HBM.
// No matrix structure -> WMMA inapplicable; optimize the data-movement path:
// 128-bit coalesced global loads/stores with non-temporal (TH=NT) hints,
// full recurrence kept in VGPRs, wave32-friendly 256-thread blocks.

typedef __attribute__((ext_vector_type(4))) float v4f;

#define IF_STEP(m, xin, sout)                    \
    do {                                         \
        (m) += (xin);                            \
        float sp_ = ((m) >= th) ? th : 0.0f;     \
        (m) -= sp_;                              \
        (sout) = sp_;                            \
    } while (0)

__global__ __launch_bounds__(256) void if_fwd_vec4(
    const float* __restrict__ x,       // [4*N] fp32, t-major
    const float* __restrict__ thresh,  // [1]
    float* __restrict__ out,           // [4*N] fp32, t-major
    unsigned int nvec,                 // N/4 float4 groups per timestep
    unsigned int N)                    // elements per timestep
{
    unsigned int i = blockIdx.x * blockDim.x + threadIdx.x;
    if (i >= nvec) return;

    const float th = thresh[0];              // uniform -> s_load_b32
    const unsigned int base = i * 4u;

    // One b128 NT load per timestep stream (streams are N floats apart).
    v4f x0 = __builtin_nontemporal_load(reinterpret_cast<const v4f*>(x + base));
    v4f x1 = __builtin_nontemporal_load(reinterpret_cast<const v4f*>(x + N + base));
    v4f x2 = __builtin_nontemporal_load(reinterpret_cast<const v4f*>(x + 2u * N + base));
    v4f x3 = __builtin_nontemporal_load(reinterpret_cast<const v4f*>(x + 3u * N + base));

    // Four independent membranes (one per packed neuron), init 0.5*thresh.
    const float mh = 0.5f * th;
    float m0 = mh, m1 = mh, m2 = mh, m3 = mh;

    v4f s0, s1, s2, s3;
    // t = 0
    IF_STEP(m0, x0.x, s0.x); IF_STEP(m1, x0.y, s0.y);
    IF_STEP(m2, x0.z, s0.z); IF_STEP(m3, x0.w, s0.w);
    // t = 1
    IF_STEP(m0, x1.x, s1.x); IF_STEP(m1, x1.y, s1.y);
    IF_STEP(m2, x1.z, s1.z); IF_STEP(m3, x1.w, s1.w);
    // t = 2
    IF_STEP(m0, x2.x, s2.x); IF_STEP(m1, x2.y, s2.y);
    IF_STEP(m2, x2.z, s2.z); IF_STEP(m3, x2.w, s2.w);
    // t = 3
    IF_STEP(m0, x3.x, s3.x); IF_STEP(m1, x3.y, s3.y);
    IF_STEP(m2, x3.z, s3.z); IF_STEP(m3, x3.w, s3.w);

    // One b128 NT store per timestep stream.
    __builtin_nontemporal_store(s0, reinterpret_cast<v4f*>(out + base));
    __builtin_nontemporal_store(s1, reinterpret_cast<v4f*>(out + N + base));
    __builtin_nontemporal_store(s2, reinterpret_cast<v4f*>(out + 2u * N + base));
    __builtin_nontemporal_store(s3, reinterpret_cast<v4f*>(out + 3u * N + base));
}

// Scalar fallback / tail (not taken for the reference shape, N % 4 == 0).
__global__ __launch_bounds__(256) void if_fwd_scalar(
    const float* __restrict__ x,
    const float* __restrict__ thresh,
    float* __restrict__ out,
    unsigned int start,
    unsigned int N)
{
    unsigned int i = start + blockIdx.x * blockDim.x + threadIdx.x;
    if (i >= N) return;
    const float th = thresh[0];
    float m = 0.5f * th;
#pragma unroll
    for (unsigned int t = 0; t < 4u; ++t) {
        float v = x[t * N + i];
        m += v;
        float s = (m >= th) ? th : 0.0f;
        m -= s;
        out[t * N + i] = s;
    }
}

extern "C" void kernel_launch(void* const* d_in, const int* in_sizes, int n_in,
                              void* d_out, int out_size, void* d_ws, size_t ws_size,
                              hipStream_t stream) {
    const float* x      = (const float*)d_in[0];
    const float* thresh = (const float*)d_in[1];
    float* out          = (float*)d_out;

    const unsigned int total = (unsigned int)in_sizes[0]; // T*N elements
    const unsigned int N     = total / 4u;                // per-timestep elements
    const unsigned int nvec  = N / 4u;                    // float4 groups
    const unsigned int rem   = N - nvec * 4u;

    if (nvec) {
        dim3 block(256);
        dim3 grid((nvec + 255u) / 256u);
        if_fwd_vec4<<<grid, block, 0, stream>>>(x, thresh, out, nvec, N);
    }
    if (rem) {
        dim3 block(256);
        dim3 grid((rem + 255u) / 256u);
        if_fwd_scalar<<<grid, block, 0, stream>>>(x, thresh, out, nvec * 4u, N);
    }
}